// LiftSplat_72267119722852
// MI455X (gfx1250) — compile-verified
//
#include <hip/hip_runtime.h>
#include <hip/hip_bf16.h>

// Problem constants (from reference)
#define B_    4
#define N_    6
#define C_    64
#define D_    41
#define FH_   16
#define FW_   44
#define NPIX  (FH_*FW_)     // 704 pixels per camera image
#define NX_   400
#define NY_   200
#define OUTN  (B_*C_*NY_*NX_)   // 20,480,000 floats

typedef __attribute__((ext_vector_type(2))) float v2f;
typedef __attribute__((ext_vector_type(8))) float v8f;

__device__ __forceinline__ void inv3x3(const float* m, float* o) {
    float a=m[0],b=m[1],c=m[2],d=m[3],e=m[4],f=m[5],g=m[6],h=m[7],i=m[8];
    float A = e*i - f*h, Bc = f*g - d*i, Cc = d*h - e*g;
    float det = a*A + b*Bc + c*Cc;
    float id = 1.0f / det;
    o[0]=A*id;          o[1]=(c*h-b*i)*id;  o[2]=(b*f-c*e)*id;
    o[3]=Bc*id;         o[4]=(a*i-c*g)*id;  o[5]=(c*d-a*f)*id;
    o[6]=Cc*id;         o[7]=(b*g-a*h)*id;  o[8]=(a*e-b*d)*id;
}

__global__ void __launch_bounds__(256)
lss_zero_out(float4* __restrict__ p, int n4) {
    int i = blockIdx.x * blockDim.x + threadIdx.x;
    int st = gridDim.x * blockDim.x;
    float4 z; z.x = z.y = z.z = z.w = 0.0f;
    for (; i < n4; i += st) p[i] = z;
}

// One wave (32 threads) per block; each block handles one 16-pixel tile of one
// camera image: full-precision WMMA GEMM -> softmax -> lift -> voxel scatter.
__global__ void __launch_bounds__(32)
lss_fused(const float* __restrict__ x,         // (24, 64, 16, 44)
          const float* __restrict__ dep_w,     // (105, 64)
          const float* __restrict__ dep_b,     // (105,)
          const float* __restrict__ rots,      // (4,6,3,3)
          const float* __restrict__ trans,     // (4,6,3)
          const float* __restrict__ intrins,   // (4,6,3,3)
          const float* __restrict__ post_rots, // (4,6,3,3)
          const float* __restrict__ post_trans,// (4,6,3)
          float* __restrict__ out)             // (4, 64, 200, 400)
{
    __shared__ float depth_s[16][D_ + 1];   // +1 pad vs bank conflicts
    __shared__ float feat_s[16][C_];

    const int tile  = blockIdx.x;          // 0..1055
    const int bn    = tile / (NPIX / 16);  // camera image 0..23
    const int pt    = tile % (NPIX / 16);
    const int p0    = pt * 16;
    const int lane  = threadIdx.x;         // 0..31 (wave32)
    const int laneP = lane & 15;
    const int hi    = lane >> 4;           // 0: K pair {0,1}; 1: K pair {2,3}

    const float* xim = x + (size_t)bn * C_ * NPIX;

    // ---- Preload all B fragments: X[k + 2*hi + {0,1}, p0 + laneP] per k-step
    v2f bf[16];
#pragma unroll
    for (int k4 = 0; k4 < 16; ++k4) {
        int k = k4 * 4 + 2 * hi;
        bf[k4].x = xim[(size_t)k       * NPIX + p0 + laneP];
        bf[k4].y = xim[(size_t)(k + 1) * NPIX + p0 + laneP];
    }

    // ---- GEMM: logits[112x16 tile] = dep_w(pad to 112x64) @ X(64x16)
    // 7 M-tiles x 16 K-steps of V_WMMA_F32_16X16X4_F32 (fp32 matrix pipe).
    v8f acc[7];
#pragma unroll
    for (int t = 0; t < 7; ++t) {
        int m = 16 * t + laneP;
        bool mok = (m < 105);
        const float* wrow = dep_w + m * C_;
        v8f c = {};
#pragma unroll
        for (int k4 = 0; k4 < 16; ++k4) {
            int k = k4 * 4 + 2 * hi;
            v2f a;
            a.x = mok ? wrow[k]     : 0.0f;
            a.y = mok ? wrow[k + 1] : 0.0f;
            c = __builtin_amdgcn_wmma_f32_16x16x4_f32(
                    false, a, false, bf[k4], (short)0, c, false, false);
        }
        acc[t] = c;
    }

    // ---- Bias. Lane owns outputs o = 16*t + 8*hi + r for its pixel column.
#pragma unroll
    for (int t = 0; t < 7; ++t)
#pragma unroll
        for (int r = 0; r < 8; ++r) {
            int o = 16 * t + 8 * hi + r;
            acc[t][r] += (o < 105) ? dep_b[o] : 0.0f;
        }

    // ---- Softmax over depth bins o in [0,41). Each pixel's logits are split
    // across lanes (p, p+16): one xor-16 shuffle combines the halves.
    float mloc = -3.0e38f;
#pragma unroll
    for (int t = 0; t < 3; ++t)
#pragma unroll
        for (int r = 0; r < 8; ++r) {
            int o = 16 * t + 8 * hi + r;
            if (o < D_) mloc = fmaxf(mloc, acc[t][r]);
        }
    float mx = fmaxf(mloc, __shfl_xor(mloc, 16, 32));
    float sloc = 0.0f;
#pragma unroll
    for (int t = 0; t < 3; ++t)
#pragma unroll
        for (int r = 0; r < 8; ++r) {
            int o = 16 * t + 8 * hi + r;
            if (o < D_) {
                float e = __expf(acc[t][r] - mx);
                acc[t][r] = e;
                sloc += e;
            }
        }
    float inv_s = 1.0f / (sloc + __shfl_xor(sloc, 16, 32));

    // ---- Spill depth / feat for the 16 pixels to LDS
#pragma unroll
    for (int t = 0; t < 7; ++t)
#pragma unroll
        for (int r = 0; r < 8; ++r) {
            int o = 16 * t + 8 * hi + r;
            if (o < D_)        depth_s[laneP][o]      = acc[t][r] * inv_s;
            else if (o < 105)  feat_s[laneP][o - D_]  = acc[t][r];
        }
    __syncthreads();

    // ---- Camera geometry (uniform per block)
    const int b = bn / N_;
    float R[9], K[9], P[9], iK[9], iP[9], Cm[9];
#pragma unroll
    for (int i = 0; i < 9; ++i) {
        R[i] = rots[bn * 9 + i];
        K[i] = intrins[bn * 9 + i];
        P[i] = post_rots[bn * 9 + i];
    }
    inv3x3(K, iK);
    inv3x3(P, iP);
#pragma unroll
    for (int i = 0; i < 3; ++i)
#pragma unroll
        for (int j = 0; j < 3; ++j)
            Cm[i*3+j] = R[i*3+0]*iK[0*3+j] + R[i*3+1]*iK[1*3+j] + R[i*3+2]*iK[2*3+j];
    const float tr0 = trans[bn*3+0], tr1 = trans[bn*3+1], tr2 = trans[bn*3+2];
    const float pt0 = post_trans[bn*3+0], pt1 = post_trans[bn*3+1], pt2 = post_trans[bn*3+2];

    // ---- Scatter: 16 pixels x 41 depth bins, 64-channel fp32 atomics (L2-resident)
    for (int j = lane; j < 16 * D_; j += 32) {
        int pl = j / D_;
        int d  = j - pl * D_;
        int p  = p0 + pl;
        int h  = p / FW_;
        int w  = p - h * FW_;
        float fx = (float)w * (703.0f / 43.0f);   // linspace(0,703,44)
        float fy = (float)h * 17.0f;              // linspace(0,255,16)
        float fd = 4.0f + (float)d;               // arange(4,45)
        // undo post-rot/trans
        float ux = fx - pt0, uy = fy - pt1, uz = fd - pt2;
        float vx = iP[0]*ux + iP[1]*uy + iP[2]*uz;
        float vy = iP[3]*ux + iP[4]*uy + iP[5]*uz;
        float vz = iP[6]*ux + iP[7]*uy + iP[8]*uz;
        // un-project: [x*z, y*z, z], then ego frame
        float qx = vx * vz, qy = vy * vz, qz = vz;
        float gx = Cm[0]*qx + Cm[1]*qy + Cm[2]*qz + tr0;
        float gy = Cm[3]*qx + Cm[4]*qy + Cm[5]*qz + tr1;
        float gz = Cm[6]*qx + Cm[7]*qy + Cm[8]*qz + tr2;
        // voxelize; C cast truncates toward zero like jnp astype(int32)
        int ix = (int)((gx + 30.0f) / 0.15f);
        int iy = (int)((gy + 15.0f) / 0.15f);
        int iz = (int)((gz + 10.0f) / 20.0f);
        if (ix >= 0 && ix < NX_ && iy >= 0 && iy < NY_ && iz == 0) {
            float dv = depth_s[pl][d];
            float* op = out + (size_t)b * C_ * (NY_ * NX_)
                            + (size_t)iy * NX_ + ix;
#pragma unroll 8
            for (int c = 0; c < C_; ++c)
                unsafeAtomicAdd(op + (size_t)c * (NY_ * NX_), dv * feat_s[pl][c]);
        }
    }
}

extern "C" void kernel_launch(void* const* d_in, const int* in_sizes, int n_in,
                              void* d_out, int out_size, void* d_ws, size_t ws_size,
                              hipStream_t stream) {
    (void)in_sizes; (void)n_in; (void)out_size; (void)d_ws; (void)ws_size;
    const float* x          = (const float*)d_in[0];
    const float* dep_w      = (const float*)d_in[1];
    const float* dep_b      = (const float*)d_in[2];
    const float* rots       = (const float*)d_in[3];
    const float* trans      = (const float*)d_in[4];
    const float* intrins    = (const float*)d_in[5];
    const float* post_rots  = (const float*)d_in[6];
    const float* post_trans = (const float*)d_in[7];
    float* out = (float*)d_out;

    // Zero the BEV grid (scatter accumulates into it).
    lss_zero_out<<<2048, 256, 0, stream>>>((float4*)out, OUTN / 4);

    // One wave per 16-pixel tile: 24 images * 44 tiles = 1056 blocks.
    lss_fused<<<(B_ * N_) * (NPIX / 16), 32, 0, stream>>>(
        x, dep_w, dep_b, rots, trans, intrins, post_rots, post_trans, out);
}